// LsqNonneg_2499670966332
// MI455X (gfx1250) — compile-verified
//
#include <hip/hip_runtime.h>
#include <math.h>

// ---------------------------------------------------------------------------
// LsqNonneg on MI455X (gfx1250): S[32,2048] = NNLS(X[256,2048], A[256,32])
//   Stage 1: AtA = A^T A, AtX = A^T X  via V_WMMA_F32_16X16X4_F32 (fp32 WMMA)
//   Stage 2: one wave32 per column; masked 32x32 solves via LDS Gauss-Jordan
// ---------------------------------------------------------------------------

typedef __attribute__((ext_vector_type(2))) float v2f;
typedef __attribute__((ext_vector_type(8))) float v8f;

#define MM 256
#define KK 32
#define NN 2048
#define TOL 1e-6f

// ---------------------------------------------------------------------------
// Stage 1a: AtA [32,32] = A^T @ A, single wave, f32 WMMA 16x16x4.
// A-fragment (16x4 MxK f32): lane L -> { A_op[L&15][2*(L>>4)], A_op[L&15][2*(L>>4)+1] }
// B-fragment (4x16 KxN f32): lane L -> { B_op[2*(L>>4)][L&15], B_op[2*(L>>4)+1][L&15] }
// For AtA both operands come from A and the two layouts read identical data.
// ---------------------------------------------------------------------------
__global__ void __launch_bounds__(32) ata_kernel(const float* __restrict__ A,
                                                 float* __restrict__ AtA) {
  const int lane = threadIdx.x & 31;
  const int i = lane & 15;     // M (and N) index within tile
  const int h = lane >> 4;     // half selector -> K rows 2h, 2h+1
  v8f c00 = {}, c01 = {}, c10 = {}, c11 = {};
  for (int m0 = 0; m0 < MM; m0 += 4) {
    v2f a0, a1;  // fragments for k-tiles [0,16) and [16,32); also serve as B
    a0.x = A[(m0 + 2 * h + 0) * KK + i];
    a0.y = A[(m0 + 2 * h + 1) * KK + i];
    a1.x = A[(m0 + 2 * h + 0) * KK + 16 + i];
    a1.y = A[(m0 + 2 * h + 1) * KK + 16 + i];
    c00 = __builtin_amdgcn_wmma_f32_16x16x4_f32(false, a0, false, a0, (short)0, c00, false, false);
    c01 = __builtin_amdgcn_wmma_f32_16x16x4_f32(false, a0, false, a1, (short)0, c01, false, false);
    c10 = __builtin_amdgcn_wmma_f32_16x16x4_f32(false, a1, false, a0, (short)0, c10, false, false);
    c11 = __builtin_amdgcn_wmma_f32_16x16x4_f32(false, a1, false, a1, (short)0, c11, false, false);
  }
  // D layout: VGPR v, lanes 0-15 -> (M=v, N=lane); lanes 16-31 -> (M=v+8, N=lane-16)
#pragma unroll
  for (int v = 0; v < 8; ++v) {
    const int r = v + 8 * h;
    AtA[(r)*KK + (i)]            = c00[v];
    AtA[(r)*KK + (16 + i)]       = c01[v];
    AtA[(16 + r) * KK + (i)]      = c10[v];
    AtA[(16 + r) * KK + (16 + i)] = c11[v];
  }
}

// ---------------------------------------------------------------------------
// Stage 1b: AtX [32,2048] = A^T @ X. 4 waves/block, one 16-wide N-tile per wave.
// ---------------------------------------------------------------------------
__global__ void __launch_bounds__(128) atx_kernel(const float* __restrict__ A,
                                                  const float* __restrict__ X,
                                                  float* __restrict__ AtX) {
  const int lane = threadIdx.x & 31;
  const int wave = threadIdx.x >> 5;
  const int n0 = (blockIdx.x * 4 + wave) * 16;
  const int i = lane & 15;
  const int h = lane >> 4;
  v8f c0 = {}, c1 = {};
#pragma unroll 4
  for (int m0 = 0; m0 < MM; m0 += 4) {
    v2f a0, a1, b;
    a0.x = A[(m0 + 2 * h + 0) * KK + i];
    a0.y = A[(m0 + 2 * h + 1) * KK + i];
    a1.x = A[(m0 + 2 * h + 0) * KK + 16 + i];
    a1.y = A[(m0 + 2 * h + 1) * KK + 16 + i];
    b.x = X[(m0 + 2 * h + 0) * NN + n0 + i];
    b.y = X[(m0 + 2 * h + 1) * NN + n0 + i];
    c0 = __builtin_amdgcn_wmma_f32_16x16x4_f32(false, a0, false, b, (short)0, c0, false, false);
    c1 = __builtin_amdgcn_wmma_f32_16x16x4_f32(false, a1, false, b, (short)0, c1, false, false);
  }
#pragma unroll
  for (int v = 0; v < 8; ++v) {
    const int r = v + 8 * h;
    AtX[(r)*NN + n0 + i]        = c0[v];
    AtX[(16 + r) * NN + n0 + i] = c1[v];
  }
}

// ---------------------------------------------------------------------------
// Wave32 reduction helpers
// ---------------------------------------------------------------------------
__device__ __forceinline__ bool wave_any(bool p) {
  int v = p ? 1 : 0;
#pragma unroll
  for (int m = 16; m; m >>= 1) v |= __shfl_xor(v, m, 32);
  return v != 0;
}
__device__ __forceinline__ float wave_min_f(float v) {
#pragma unroll
  for (int m = 16; m; m >>= 1) v = fminf(v, __shfl_xor(v, m, 32));
  return v;
}
__device__ __forceinline__ float wave_max_f(float v) {
#pragma unroll
  for (int m = 16; m; m >>= 1) v = fmaxf(v, __shfl_xor(v, m, 32));
  return v;
}
__device__ __forceinline__ int wave_min_i(int v) {
#pragma unroll
  for (int m = 16; m; m >>= 1) v = min(v, __shfl_xor(v, m, 32));
  return v;
}

// ---------------------------------------------------------------------------
// Masked solve: (AtA*(P x P) + diag(1-P)) z = Atb*P  via wave-parallel
// Gauss-Jordan; lane r owns row r of the 32x33 augmented system in LDS.
// Stride 33 (coprime with 64 banks) -> conflict-free row accesses.
// ---------------------------------------------------------------------------
__device__ float masked_solve(const float* __restrict__ ldsAtA,
                              float* __restrict__ msys,
                              int r, float Pm, float Atb_r) {
#pragma unroll
  for (int j = 0; j < 32; ++j) {
    float Pmj = __shfl(Pm, j, 32);
    float v = ldsAtA[r * 33 + j] * (Pm * Pmj);
    v = (j == r) ? (v + 1.0f - Pm) : v;
    msys[r * 33 + j] = v;
  }
  msys[r * 33 + 32] = Atb_r * Pm;
  __builtin_amdgcn_wave_barrier();
  for (int piv = 0; piv < 32; ++piv) {
    float invd = 1.0f / msys[piv * 33 + piv];  // broadcast LDS read
    float f = msys[r * 33 + piv] * invd;
    f = (r == piv) ? 0.0f : f;
    for (int j = piv; j < 33; ++j) {
      msys[r * 33 + j] -= f * msys[piv * 33 + j];
    }
    __builtin_amdgcn_wave_barrier();
  }
  float z = msys[r * 33 + 32] / msys[r * 33 + r];
  return z * Pm;
}

// ---------------------------------------------------------------------------
// Stage 2: active-set NNLS, one wave per column (lane r = coefficient r).
// ---------------------------------------------------------------------------
__global__ void __launch_bounds__(256) nnls_kernel(const float* __restrict__ AtA,
                                                   const float* __restrict__ AtX,
                                                   float* __restrict__ S) {
  __shared__ float ldsAtA[32 * 33];
  __shared__ float ldsMsys[8 * 32 * 33];

  const int tid = threadIdx.x;
  const int lane = tid & 31;
  const int wave = tid >> 5;
  const int col = blockIdx.x * 8 + wave;

  // Cache AtA once per workgroup (shared by all 8 column-solver waves).
  for (int e = tid; e < 32 * 32; e += 256)
    ldsAtA[(e >> 5) * 33 + (e & 31)] = AtA[e];
  __syncthreads();

  float* msys = &ldsMsys[wave * 32 * 33];
  const int r = lane;
  const float Atb = AtX[r * NN + col];

  bool P = false;
  float s = 0.0f, z = 0.0f;

  for (int it = 0; it < 3 * KK; ++it) {
    // w = Atb - AtA @ s  (s_j gathered via lane shuffles)
    float dot = 0.0f;
#pragma unroll
    for (int j = 0; j < 32; ++j)
      dot = fmaf(ldsAtA[r * 33 + j], __shfl(s, j, 32), dot);
    const float wfull = Atb - dot;
    if (!wave_any(!P && wfull > TOL)) break;

    // argmax over passive set (ties -> lowest index, like jnp.argmax)
    const float w = P ? -__builtin_inff() : wfull;
    const float wmax = wave_max_f(w);
    int idx = (w == wmax) ? r : 64;
    idx = wave_min_i(idx);
    P = P || (r == idx);

    z = masked_solve(ldsAtA, msys, r, P ? 1.0f : 0.0f, Atb);

    // inner loop: shrink active set until z feasible
    for (int in_it = 0; in_it < 3 * KK; ++in_it) {
      if (!wave_any(P && z <= TOL)) break;
      const bool mask = P && (z <= TOL);
      const float denom = s - z;
      const float ratio =
          mask ? (s / ((denom != 0.0f) ? denom : 1.0f)) : __builtin_inff();
      const float alpha = wave_min_f(ratio);
      s = s + alpha * (z - s);
      P = P && (s > TOL);
      s = P ? s : 0.0f;
      z = masked_solve(ldsAtA, msys, r, P ? 1.0f : 0.0f, Atb);
    }
    s = P ? z : 0.0f;
  }

  S[r * NN + col] = s;
}

// ---------------------------------------------------------------------------
extern "C" void kernel_launch(void* const* d_in, const int* in_sizes, int n_in,
                              void* d_out, int out_size, void* d_ws, size_t ws_size,
                              hipStream_t stream) {
  const float* X = (const float*)d_in[0];  // [256, 2048]
  const float* A = (const float*)d_in[1];  // [256, 32]
  float* S = (float*)d_out;                // [32, 2048]

  float* AtX = (float*)d_ws;               // 32*2048 floats = 256 KB
  float* AtA = AtX + KK * NN;              // 32*32 floats  = 4 KB

  ata_kernel<<<1, 32, 0, stream>>>(A, AtA);
  atx_kernel<<<NN / (16 * 4), 128, 0, stream>>>(A, X, AtX);
  nnls_kernel<<<NN / 8, 256, 0, stream>>>(AtA, AtX, S);
}